// GPhyRLayer_19284403159776
// MI455X (gfx1250) — compile-verified
//
#include <hip/hip_runtime.h>
#include <stdint.h>
#include <math.h>

// ---------------------------------------------------------------------------
// Gumbel-perturbed Kruskal MST (matches JAX reference):
//   scores = sigmoid(s) + (-log(-log(u+eps)+eps))
//   stable argsort descending, serial union-find (union by rank, no path
//   compression), b[idx] = 1 for accepted edges (at most num_nodes-1).
//
// GPU plan: elementwise key build -> bitonic sort of packed u64 keys
// (descending score, ascending index on ties == stable argsort(-scores))
// -> single-thread Kruskal scan.  The bitonic inner passes (j <= 1024) run in
// LDS; tiles are staged with CDNA5 async global->LDS DMA (ASYNCcnt path).
// ---------------------------------------------------------------------------

#define TILE 2048u          // keys per LDS tile (16 KiB of LDS)
#define LDS_THREADS 1024u   // one compare/exchange pair per thread

// ----- CDNA5 async global<->LDS path (guarded so the file always compiles) --
#if defined(__HIP_DEVICE_COMPILE__)
#  ifdef __has_builtin
#    if __has_builtin(__builtin_amdgcn_global_load_async_to_lds_b128)
#      define USE_ASYNC_LOAD 1
#    endif
#    if __has_builtin(__builtin_amdgcn_global_store_async_from_lds_b128)
#      define USE_ASYNC_STORE 1
#    endif
#  endif
#endif

#if defined(USE_ASYNC_LOAD) || defined(USE_ASYNC_STORE)
// Builtin expects pointer-to-int4: param0 in global (AS1), param1 LDS (AS3).
typedef int v4i __attribute__((ext_vector_type(4)));
typedef __attribute__((address_space(1))) v4i glob_v4i;
typedef __attribute__((address_space(3))) v4i lds_v4i;
__device__ __forceinline__ void wait_asynccnt0() {
#  ifdef __has_builtin
#    if __has_builtin(__builtin_amdgcn_s_wait_asynccnt)
  __builtin_amdgcn_s_wait_asynccnt(0);
#    else
  asm volatile("s_wait_asynccnt 0" ::: "memory");
#    endif
#  else
  asm volatile("s_wait_asynccnt 0" ::: "memory");
#  endif
}
#endif

// Map float -> u32 such that ascending u32 order == DESCENDING float order.
__device__ __forceinline__ uint32_t float_key_desc(float f) {
  uint32_t x = __float_as_uint(f);
  x = (x & 0x80000000u) ? ~x : (x | 0x80000000u);  // ascending order
  return ~x;                                       // flip -> descending
}

// ---------------------------------------------------------------------------
// Kernel 1: build sort keys, zero output, (re)init union-find state.
// ---------------------------------------------------------------------------
__global__ void gphyr_init(const float* __restrict__ s,
                           const float* __restrict__ u,
                           float* __restrict__ out,
                           uint64_t* __restrict__ keys,
                           int* __restrict__ parent,
                           int* __restrict__ rnk,
                           const int* __restrict__ nptr,
                           int E, int P, int NMAX) {
  int i = blockIdx.x * blockDim.x + threadIdx.x;
  if (i < P) {
    if (i < E) {
      float sv = s[i];
      float uv = u[i];
      float sp = 1.0f / (1.0f + expf(-sv));
      float g  = -logf(-logf(uv + 1e-9f) + 1e-9f);
      float sc = sp + g;
      keys[i] = ((uint64_t)float_key_desc(sc) << 32) | (uint32_t)i;
      out[i]  = 0.0f;
    } else {
      keys[i] = ~0ull;  // padding sorts to the end (idx field = 0xFFFFFFFF)
    }
  }
  int n = nptr[0];
  if (n > NMAX) n = NMAX;
  if (i < n) {
    parent[i] = i;
    rnk[i]    = 0;
  }
}

// ---------------------------------------------------------------------------
// Kernel 2: one global-memory bitonic compare/exchange pass (j > TILE/2).
// ---------------------------------------------------------------------------
__global__ void gphyr_bitonic_global(uint64_t* __restrict__ keys,
                                     uint32_t k, uint32_t j, uint32_t half) {
  uint32_t t = blockIdx.x * blockDim.x + threadIdx.x;
  if (t >= half) return;
  uint32_t i = ((t & ~(j - 1u)) << 1) | (t & (j - 1u));
  uint32_t p = i | j;
  uint64_t a = keys[i];
  uint64_t b = keys[p];
  bool asc = ((i & k) == 0u);
  bool sw  = asc ? (a > b) : (a < b);
  if (sw) { keys[i] = b; keys[p] = a; }
}

// ---------------------------------------------------------------------------
// Kernel 3: all bitonic passes with j <= TILE/2, tile resident in LDS.
// Tiles staged via CDNA5 async global->LDS DMA when the builtins exist.
// ---------------------------------------------------------------------------
__global__ __launch_bounds__(LDS_THREADS)
void gphyr_bitonic_lds(uint64_t* __restrict__ keys, uint32_t k, uint32_t jstart) {
  __shared__ uint64_t tile[TILE];
  uint32_t t    = threadIdx.x;
  uint32_t base = blockIdx.x * TILE;
  uint32_t li   = 2u * t;  // this thread owns tile[li], tile[li+1]

#ifdef USE_ASYNC_LOAD
  // 16 bytes (two u64 keys) per lane, DMA'd straight into LDS (ASYNCcnt).
  __builtin_amdgcn_global_load_async_to_lds_b128(
      (glob_v4i*)(keys + base + li), (lds_v4i*)&tile[li], 0, 0);
  wait_asynccnt0();
#else
  tile[li]      = keys[base + li];
  tile[li + 1u] = keys[base + li + 1u];
#endif
  __syncthreads();

  for (uint32_t j = jstart; j >= 1u; j >>= 1) {
    uint32_t i = ((t & ~(j - 1u)) << 1) | (t & (j - 1u));
    uint32_t p = i | j;
    uint64_t a = tile[i];
    uint64_t b = tile[p];
    bool asc = (((base + i) & k) == 0u);
    bool sw  = asc ? (a > b) : (a < b);
    if (sw) { tile[i] = b; tile[p] = a; }
    __syncthreads();
  }

#ifdef USE_ASYNC_STORE
  __builtin_amdgcn_global_store_async_from_lds_b128(
      (glob_v4i*)(keys + base + li), (lds_v4i*)&tile[li], 0, 0);
  // S_ENDPGM performs an implicit wait-idle; completion ordered vs next kernel.
#else
  keys[base + li]      = tile[li];
  keys[base + li + 1u] = tile[li + 1u];
#endif
}

// ---------------------------------------------------------------------------
// Kernel 4: serial Kruskal (inherently sequential; exact reference semantics).
// ---------------------------------------------------------------------------
__global__ void gphyr_kruskal(const uint64_t* __restrict__ keys,
                              const int* __restrict__ edge_index,
                              float* __restrict__ out,
                              int* __restrict__ parent,
                              int* __restrict__ rnk,
                              const int* __restrict__ nptr,
                              int E, int P, int NMAX) {
  if (blockIdx.x != 0 || threadIdx.x != 0) return;
  const int* src = edge_index;
  const int* dst = edge_index + E;
  int n = nptr[0];
  if (n > NMAX) n = NMAX;
  int maxu = n - 1;
  int used = 0;
  for (int e = 0; e < P && used < maxu; ++e) {
    uint32_t idx = (uint32_t)keys[e];
    if (idx >= (uint32_t)E) continue;  // padding entry
    int a = src[idx];
    int b = dst[idx];
    int ru = a; while (parent[ru] != ru) ru = parent[ru];
    int rv = b; while (parent[rv] != rv) rv = parent[rv];
    if (ru != rv) {
      int ri = rnk[ru];
      int rj = rnk[rv];
      int child = (ri < rj) ? ru : rv;
      int root  = (ri < rj) ? rv : ru;
      parent[child] = root;
      if (ri == rj) rnk[root] = ri + 1;  // root==ru on ties, as in reference
      out[idx] = 1.0f;
      ++used;
    }
  }
}

// ---------------------------------------------------------------------------
// Host launcher (graph-capture safe: launches only, all on `stream`).
// ---------------------------------------------------------------------------
extern "C" void kernel_launch(void* const* d_in, const int* in_sizes, int n_in,
                              void* d_out, int out_size, void* d_ws, size_t ws_size,
                              hipStream_t stream) {
  const float* s    = (const float*)d_in[0];
  const float* u    = (const float*)d_in[1];
  const int*   ei   = (const int*)d_in[2];   // (2, E) row-major: src then dst
  const int*   nptr = (const int*)d_in[3];   // num_nodes (device scalar)
  float* out = (float*)d_out;

  const int E = in_sizes[0];
  int P = 1;
  while (P < E) P <<= 1;                      // 400000 -> 524288 = 2^19
  const int NMAX = 100000;                    // NUM_NODES in the reference

  // Workspace layout: keys (P u64) | parent (NMAX i32) | rank (NMAX i32)
  char* ws = (char*)d_ws;
  uint64_t* keys = (uint64_t*)ws;
  size_t off = (size_t)P * sizeof(uint64_t);
  int* parent = (int*)(ws + off);
  off += ((size_t)NMAX * sizeof(int) + 255) & ~(size_t)255;
  int* rnk = (int*)(ws + off);
  (void)ws_size; (void)n_in; (void)out_size;

  // 1) keys + output zero + union-find init
  {
    int blocks = (P + 255) / 256;
    gphyr_init<<<blocks, 256, 0, stream>>>(s, u, out, keys, parent, rnk,
                                           nptr, E, P, NMAX);
  }

  // 2) bitonic sort of P u64 keys (ascending == stable argsort(-scores))
  const uint32_t half = (uint32_t)(P >> 1);
  const uint32_t gblocks = (half + 255u) / 256u;
  const uint32_t lblocks = (uint32_t)P / TILE;
  for (uint32_t k = 2; k <= (uint32_t)P; k <<= 1) {
    uint32_t j = k >> 1;
    for (; j > TILE / 2u; j >>= 1)
      gphyr_bitonic_global<<<gblocks, 256, 0, stream>>>(keys, k, j, half);
    gphyr_bitonic_lds<<<lblocks, LDS_THREADS, 0, stream>>>(keys, k, j);
  }

  // 3) serial Kruskal scan
  gphyr_kruskal<<<1, 1, 0, stream>>>(keys, ei, out, parent, rnk,
                                     nptr, E, P, NMAX);
}